// IVA_NG_37340445671456
// MI455X (gfx1250) — compile-verified
//
#include <hip/hip_runtime.h>
#include <hip/hip_bf16.h>

// Problem constants (from reference)
#define N_MIC   4
#define N_COMP  4
#define JJ      3000
#define II      513
#define NSTEPS  40        // N_ITER * N_COMPONENTS
#define LR_C    0.1f
#define IP_EPS  1e-20f

// TDM chunking for the cov kernel: 25 chunks of 120 j-samples
#define CHUNK   120
#define NCHUNK  25

typedef float v2f __attribute__((ext_vector_type(2)));
typedef float v8f __attribute__((ext_vector_type(8)));
typedef unsigned int v4u __attribute__((ext_vector_type(4)));
typedef int v4i __attribute__((ext_vector_type(4)));
typedef int v8i __attribute__((ext_vector_type(8)));

// ---------------------------------------------------------------------------
// Setup: transpose X (M,J,I,2) -> planar Xre/Xim [i][m][j]  (j contiguous)
// ---------------------------------------------------------------------------
__global__ void iva_transpose_kernel(const float* __restrict__ X,
                                     float* __restrict__ Xre,
                                     float* __restrict__ Xim) {
    int t = blockIdx.x * blockDim.x + threadIdx.x;
    const int total = N_MIC * JJ * II;
    if (t >= total) return;
    int i  = t % II;
    int mj = t / II;
    int j  = mj % JJ;
    int m  = mj / JJ;
    float xr = X[(size_t)t * 2 + 0];
    float xi = X[(size_t)t * 2 + 1];
    size_t d = (size_t)(i * N_MIC + m) * JJ + j;
    Xre[d] = xr;
    Xim[d] = xi;
}

// W[i] = complex identity
__global__ void iva_init_w_kernel(float* __restrict__ Wre, float* __restrict__ Wim) {
    int t = blockIdx.x * blockDim.x + threadIdx.x;
    if (t >= II * 16) return;
    int e = t & 15;
    Wre[t] = ((e >> 2) == (e & 3)) ? 1.0f : 0.0f;
    Wim[t] = 0.0f;
}

// ---------------------------------------------------------------------------
// Pass 1 per step: sm[n][j] = sum_i |Y[i,n,j]|^2 ; one thread owns one j.
// All 513 W matrices staged once into LDS (65.6 KB); the i-loop then runs on
// DS + the coalesced j-contiguous X stream (L2-resident, 49 MB/step).
// ---------------------------------------------------------------------------
__global__ __launch_bounds__(256) void iva_sm_kernel(
    const float* __restrict__ Xre, const float* __restrict__ Xim,
    const float* __restrict__ Wre, const float* __restrict__ Wim,
    float* __restrict__ sm) {
    __shared__ float sWre[II * 16];
    __shared__ float sWim[II * 16];
    for (int t = threadIdx.x; t < II * 16; t += 256) {
        sWre[t] = Wre[t];
        sWim[t] = Wim[t];
    }
    __syncthreads();

    int j = blockIdx.x * blockDim.x + threadIdx.x;
    if (j >= JJ) return;
    float acc[N_COMP] = {0.f, 0.f, 0.f, 0.f};
    for (int i = 0; i < II; ++i) {
        float xr[N_MIC], xi[N_MIC];
#pragma unroll
        for (int m = 0; m < N_MIC; ++m) {
            size_t s = (size_t)(i * N_MIC + m) * JJ + j;
            xr[m] = Xre[s];
            xi[m] = Xim[s];
        }
        const float* wre = sWre + i * 16;
        const float* wim = sWim + i * 16;
#pragma unroll
        for (int n = 0; n < N_COMP; ++n) {
            float yre = 0.f, yim = 0.f;
#pragma unroll
            for (int m = 0; m < N_MIC; ++m) {
                float wr = wre[n * 4 + m], wi2 = wim[n * 4 + m];
                yre += wr * xr[m] - wi2 * xi[m];
                yim += wr * xi[m] + wi2 * xr[m];
            }
            acc[n] += yre * yre + yim * yim;
        }
    }
#pragma unroll
    for (int n = 0; n < N_COMP; ++n) sm[n * JJ + j] = acc[n];
}

// Pass 2: ic = 1 / (sqrt(sm) + eps)
__global__ void iva_ic_kernel(const float* __restrict__ sm, float* __restrict__ ic) {
    int t = blockIdx.x * blockDim.x + threadIdx.x;
    if (t >= N_COMP * JJ) return;
    ic[t] = 1.0f / (sqrtf(sm[t]) + IP_EPS);
}

// ---------------------------------------------------------------------------
// TDM: DMA a 4(m) x CHUNK(j) f32 tile (row stride JJ) from global into
// contiguous LDS.  D# built per CDNA5 ISA:
//   group0: count=1 | lds_addr | global_addr(57b) | type=2
//   group1: data_size=4B, tensor_dim0=CHUNK, tensor_dim1=4,
//           tile_dim0=CHUNK, tile_dim1=4, tensor_dim0_stride=JJ
// This toolchain's builtin is the 6-arg form:
//   (v4u g0, v8i g1, v4i g2, v4i g3, v8i g4, i32 cpol)
// ---------------------------------------------------------------------------
__device__ __forceinline__ void tdm_tile_load(const float* gsrc, unsigned lds_off) {
    unsigned long long ga = (unsigned long long)(uintptr_t)gsrc;
    union { unsigned u[4]; v4u v; } g0;
    g0.u[0] = 1u;                                               // count=1, no gather
    g0.u[1] = lds_off;                                          // lds_addr (bytes)
    g0.u[2] = (unsigned)ga;                                     // global_addr[31:0]
    g0.u[3] = ((unsigned)(ga >> 32) & 0x01FFFFFFu) | (2u << 30); // addr[56:32] | type=2
    union { int i[8]; v8i v; } g1;
    g1.i[0] = (int)(2u << 16);                    // wg_mask=0, data_size=2 (4 bytes)
    g1.i[1] = (int)((unsigned)CHUNK << 16);       // tensor_dim0[15:0] << 16
    g1.i[2] = (int)(4u << 16);                    // tensor_dim0 hi=0 | tensor_dim1=4
    g1.i[3] = (int)((unsigned)CHUNK << 16);       // tensor_dim1 hi=0 | tile_dim0=CHUNK
    g1.i[4] = 4;                                  // tile_dim1=4, tile_dim2=0
    g1.i[5] = JJ;                                 // tensor_dim0_stride[31:0]
    g1.i[6] = 0;                                  // stride0 hi | stride1 lo
    g1.i[7] = 0;                                  // stride1 hi
    v4i z4 = {0, 0, 0, 0};
    v8i z8 = {0, 0, 0, 0, 0, 0, 0, 0};
    __builtin_amdgcn_tensor_load_to_lds(g0.v, g1.v, z4, z4, z8, 0);
}

// ---------------------------------------------------------------------------
// Pass 3: covariance via V_WMMA_F32_16X16X4_F32 (two frequencies per wave)
// and W natural-gradient update.  X tiles are staged by the Tensor Data
// Mover into double-buffered LDS (TENSORcnt pipelined), since a single
// wave32 cannot hide VMEM latency on its own.
//
// Real 8x8 Gram S_f = P_f * Q_f^T (contracted over j) with
//   P rows 0..3 = phi_re[n], 4..7 = phi_im[n];  Q rows = [Y_re ; Y_im]
// A[16x4]: rows 0-7 from f0, 8-15 from f1 at 4 consecutive j (K-dim)
// B[4x16]: cols 0-7 from f0, 8-15 from f1 (same j's)
// Diagonal 8x8 blocks of D are S_f0 / S_f1; cross blocks ignored.
// ---------------------------------------------------------------------------
__global__ __launch_bounds__(32) void iva_cov_w_kernel(
    const float* __restrict__ Xre, const float* __restrict__ Xim,
    float* __restrict__ Wre, float* __restrict__ Wim,
    const float* __restrict__ ic) {
    __shared__ float lds_x[2][2][2][N_MIC][CHUNK];  // [buf][fsel][plane][m][j] 15 KB
    __shared__ float s_cre[2][4][4];                // [fsel][n][m]
    __shared__ float s_cim[2][4][4];

    const int lane = threadIdx.x;
    const int f0 = 2 * blockIdx.x;
    int f1 = f0 + 1;
    if (f1 >= II) f1 = II - 1;  // duplicate last odd freq (benign identical writes)
    const int fpair[2] = {f0, f1};

    const int g    = lane >> 4;   // K-group: 0 -> j{0,1}, 1 -> j{2,3}
    const int rowM = lane & 15;   // A-row == B-col for this lane
    const int fsel = rowM >> 3;
    const int freq = fsel ? f1 : f0;
    const int r    = rowM & 7;
    const int n    = r & 3;
    const int isIm = r >> 2;

    // this lane's W row (fixed within a step)
    float wr[4], wi[4];
#pragma unroll
    for (int m = 0; m < 4; ++m) {
        wr[m] = Wre[freq * 16 + n * 4 + m];
        wi[m] = Wim[freq * 16 + n * 4 + m];
    }
    const float* icn = ic + n * JJ;

    v8f c = {0.f, 0.f, 0.f, 0.f, 0.f, 0.f, 0.f, 0.f};

    // prologue: TDM chunk 0 into buffer 0 (4 tile DMAs: 2 freqs x {re,im})
#pragma unroll
    for (int fq = 0; fq < 2; ++fq) {
        const float* bre = Xre + (size_t)fpair[fq] * (N_MIC * JJ);
        const float* bim = Xim + (size_t)fpair[fq] * (N_MIC * JJ);
        tdm_tile_load(bre, (unsigned)(uintptr_t)&lds_x[0][fq][0][0][0]);
        tdm_tile_load(bim, (unsigned)(uintptr_t)&lds_x[0][fq][1][0][0]);
    }

    for (int cb = 0; cb < NCHUNK; ++cb) {
        const int cur = cb & 1;
        if (cb + 1 < NCHUNK) {
            // issue next chunk into the other buffer, then wait for current:
            // in-order TENSORcnt completion => <=4 outstanding == current done
            const int jb = (cb + 1) * CHUNK;
#pragma unroll
            for (int fq = 0; fq < 2; ++fq) {
                const float* bre = Xre + (size_t)fpair[fq] * (N_MIC * JJ) + jb;
                const float* bim = Xim + (size_t)fpair[fq] * (N_MIC * JJ) + jb;
                tdm_tile_load(bre, (unsigned)(uintptr_t)&lds_x[cur ^ 1][fq][0][0][0]);
                tdm_tile_load(bim, (unsigned)(uintptr_t)&lds_x[cur ^ 1][fq][1][0][0]);
            }
            __builtin_amdgcn_s_wait_tensorcnt(4);
        } else {
            __builtin_amdgcn_s_wait_tensorcnt(0);
        }
        __syncthreads();  // TDM->LDS visible to all lanes before DS reads

        const int jglob0 = cb * CHUNK;
#pragma unroll 2
        for (int jb2 = 0; jb2 < CHUNK; jb2 += 4) {
            int jl0 = jb2 + 2 * g;          // this lane's local K columns
            float v[2];
#pragma unroll
            for (int t = 0; t < 2; ++t) {
                int jl = jl0 + t;
                float yre = 0.f, yim = 0.f;
#pragma unroll
                for (int m = 0; m < 4; ++m) {
                    float xr = lds_x[cur][fsel][0][m][jl];
                    float xi = lds_x[cur][fsel][1][m][jl];
                    yre += wr[m] * xr - wi[m] * xi;
                    yim += wr[m] * xi + wi[m] * xr;
                }
                v[t] = isIm ? yim : yre;
            }
            v2f a, b;
            a.x = v[0] * icn[jglob0 + jl0];      // phi component (ic depends on n only)
            a.y = v[1] * icn[jglob0 + jl0 + 1];
            b.x = v[0];                          // plain Y (conj folded into signs below)
            b.y = v[1];
            c = __builtin_amdgcn_wmma_f32_16x16x4_f32(
                /*neg_a=*/false, a, /*neg_b=*/false, b,
                /*c_mod=*/(short)0, c, /*reuse_a=*/false, /*reuse_b=*/false);
        }
        __syncthreads();  // all DS reads of 'cur' done before TDM overwrites it
    }

    // Recombine real Gram blocks into complex cov.
    // Useful lanes: f0 cols at lanes 0-7, f1 cols at lanes 24-31; lane^4: m <-> m+4.
    const float invJ = 1.0f / (float)JJ;
    float cre[4], cim[4];
#pragma unroll
    for (int q = 0; q < 4; ++q) {
        float p_hi = __shfl_xor(c[q + 4], 4);  // S[q+4, col^4]
        float p_lo = __shfl_xor(c[q],     4);  // S[q,   col^4]
        cre[q] = (c[q]     + p_hi) * invJ;     // S[n,m] + S[n+4,m+4]
        cim[q] = (c[q + 4] - p_lo) * invJ;     // S[n+4,m] - S[n,m+4]
    }
    bool wf0 = (lane < 4);
    bool wf1 = (lane >= 24 && lane < 28);
    if (wf0 || wf1) {
        int fi = wf1 ? 1 : 0;
        int m = lane & 3;
#pragma unroll
        for (int q = 0; q < 4; ++q) {
            s_cre[fi][q][m] = cre[q];
            s_cim[fi][q][m] = cim[q];
        }
    }
    __syncthreads();

    // W += LR * ((I - cov) @ W) ; one lane per output element, 2 freqs x 16
    {
        int fu = lane >> 4;
        int fq = fu ? f1 : f0;
        int n2 = (lane >> 2) & 3;
        int m2 = lane & 3;
        float accre = 0.f, accim = 0.f;
#pragma unroll
        for (int k = 0; k < 4; ++k) {
            float tre = ((n2 == k) ? 1.0f : 0.0f) - s_cre[fu][n2][k];
            float tim = -s_cim[fu][n2][k];
            float wkr = Wre[fq * 16 + k * 4 + m2];
            float wki = Wim[fq * 16 + k * 4 + m2];
            accre += tre * wkr - tim * wki;
            accim += tre * wki + tim * wkr;
        }
        float wnr = Wre[fq * 16 + n2 * 4 + m2];
        float wni = Wim[fq * 16 + n2 * 4 + m2];
        __syncthreads();  // all old-W reads complete before any write
        Wre[fq * 16 + n2 * 4 + m2] = wnr + LR_C * accre;
        Wim[fq * 16 + n2 * 4 + m2] = wni + LR_C * accim;
    }
}

// ---------------------------------------------------------------------------
// Finale A: scale[i][n] = inv(W[i])[REF_MIC-1=0][n]  (4x4 complex Gauss-Jordan)
// ---------------------------------------------------------------------------
__global__ void iva_scale_kernel(const float* __restrict__ Wre,
                                 const float* __restrict__ Wim,
                                 float* __restrict__ sre,
                                 float* __restrict__ sim) {
    int i = blockIdx.x * blockDim.x + threadIdx.x;
    if (i >= II) return;
    float ar[4][8], ai[4][8];
    for (int nn = 0; nn < 4; ++nn)
        for (int mm = 0; mm < 8; ++mm) {
            if (mm < 4) {
                ar[nn][mm] = Wre[i * 16 + nn * 4 + mm];
                ai[nn][mm] = Wim[i * 16 + nn * 4 + mm];
            } else {
                ar[nn][mm] = (nn == (mm - 4)) ? 1.0f : 0.0f;
                ai[nn][mm] = 0.0f;
            }
        }
    for (int col = 0; col < 4; ++col) {
        int p = col;
        float best = ar[col][col] * ar[col][col] + ai[col][col] * ai[col][col];
        for (int rr = col + 1; rr < 4; ++rr) {
            float mg = ar[rr][col] * ar[rr][col] + ai[rr][col] * ai[rr][col];
            if (mg > best) { best = mg; p = rr; }
        }
        if (p != col)
            for (int cc = 0; cc < 8; ++cc) {
                float tr = ar[col][cc]; ar[col][cc] = ar[p][cc]; ar[p][cc] = tr;
                float ti = ai[col][cc]; ai[col][cc] = ai[p][cc]; ai[p][cc] = ti;
            }
        float pr = ar[col][col], pi = ai[col][col];
        float dd = pr * pr + pi * pi + 1e-30f;
        float inr = pr / dd, ini = -pi / dd;
        for (int cc = 0; cc < 8; ++cc) {
            float r0 = ar[col][cc], i0 = ai[col][cc];
            ar[col][cc] = r0 * inr - i0 * ini;
            ai[col][cc] = r0 * ini + i0 * inr;
        }
        for (int rr = 0; rr < 4; ++rr) {
            if (rr == col) continue;
            float fr = ar[rr][col], fi = ai[rr][col];
            for (int cc = 0; cc < 8; ++cc) {
                float cr = ar[col][cc], cii = ai[col][cc];
                ar[rr][cc] -= fr * cr - fi * cii;
                ai[rr][cc] -= fr * cii + fi * cr;
            }
        }
    }
    for (int nn = 0; nn < 4; ++nn) {
        sre[i * 4 + nn] = ar[0][4 + nn];  // row REF_MIC-1 = 0 of inverse
        sim[i * 4 + nn] = ai[0][4 + nn];
    }
}

// ---------------------------------------------------------------------------
// Finale B: out[n][j][i][{re,im}] = scale[i][n] * (W[i] @ Xc[i,:,j])[n]
// Reads original X layout (i contiguous) so reads + writes are coalesced.
// ---------------------------------------------------------------------------
__global__ void iva_out_kernel(const float* __restrict__ X,
                               const float* __restrict__ Wre,
                               const float* __restrict__ Wim,
                               const float* __restrict__ sre,
                               const float* __restrict__ sim,
                               float* __restrict__ out) {
    int t = blockIdx.x * blockDim.x + threadIdx.x;
    if (t >= JJ * II) return;
    int i = t % II;
    int j = t / II;
    float xr[N_MIC], xi[N_MIC];
#pragma unroll
    for (int m = 0; m < N_MIC; ++m) {
        size_t s = ((size_t)(m * JJ + j) * II + i) * 2;
        xr[m] = X[s];
        xi[m] = X[s + 1];
    }
#pragma unroll
    for (int nn = 0; nn < N_COMP; ++nn) {
        float yre = 0.f, yim = 0.f;
#pragma unroll
        for (int m = 0; m < N_MIC; ++m) {
            float wr = Wre[i * 16 + nn * 4 + m];
            float wi2 = Wim[i * 16 + nn * 4 + m];
            yre += wr * xr[m] - wi2 * xi[m];
            yim += wr * xi[m] + wi2 * xr[m];
        }
        float scr = sre[i * 4 + nn], sci = sim[i * 4 + nn];
        size_t d = ((size_t)(nn * JJ + j) * II + i) * 2;
        out[d]     = scr * yre - sci * yim;
        out[d + 1] = scr * yim + sci * yre;
    }
}

// ---------------------------------------------------------------------------
extern "C" void kernel_launch(void* const* d_in, const int* in_sizes, int n_in,
                              void* d_out, int out_size, void* d_ws, size_t ws_size,
                              hipStream_t stream) {
    const float* X = (const float*)d_in[0];
    float* out = (float*)d_out;

    // workspace carve-out (floats): ~49.4 MB total
    const size_t IMJ = (size_t)II * N_MIC * JJ;  // 6,156,000
    float* ws  = (float*)d_ws;
    size_t off = 0;
    float* Xre = ws + off; off += IMJ;
    float* Xim = ws + off; off += IMJ;
    float* Wre = ws + off; off += (size_t)II * 16;
    float* Wim = ws + off; off += (size_t)II * 16;
    float* sm  = ws + off; off += (size_t)N_COMP * JJ;
    float* ic  = ws + off; off += (size_t)N_COMP * JJ;
    float* sre = ws + off; off += (size_t)II * N_COMP;
    float* sim = ws + off; off += (size_t)II * N_COMP;
    (void)ws_size; (void)in_sizes; (void)n_in; (void)out_size;

    const int BT = 256;
    const int totalX = N_MIC * JJ * II;
    iva_transpose_kernel<<<(totalX + BT - 1) / BT, BT, 0, stream>>>(X, Xre, Xim);
    iva_init_w_kernel<<<(II * 16 + BT - 1) / BT, BT, 0, stream>>>(Wre, Wim);

    const int covBlocks = (II + 1) / 2;  // 257 waves, 2 freqs each
    for (int s = 0; s < NSTEPS; ++s) {
        iva_sm_kernel<<<(JJ + BT - 1) / BT, BT, 0, stream>>>(Xre, Xim, Wre, Wim, sm);
        iva_ic_kernel<<<(N_COMP * JJ + BT - 1) / BT, BT, 0, stream>>>(sm, ic);
        iva_cov_w_kernel<<<covBlocks, 32, 0, stream>>>(Xre, Xim, Wre, Wim, ic);
    }

    iva_scale_kernel<<<(II + BT - 1) / BT, BT, 0, stream>>>(Wre, Wim, sre, sim);
    iva_out_kernel<<<((JJ * II) + BT - 1) / BT, BT, 0, stream>>>(X, Wre, Wim, sre, sim, out);
}